// Quantumnet_11390253269310
// MI455X (gfx1250) — compile-verified
//
#include <hip/hip_runtime.h>
#include <math.h>

// ---------------------------------------------------------------------------
// Fused Quantumnet for MI455X (gfx1250, wave32):
//   out[b,:] = postW @ qcircuit(tanh(input[b,:] @ preW^T + preb) * pi/2) + postb
//
// Roofline: 128MB activation read + 26MB output write at 23.3 TB/s ~= 6.6us.
// The pipeline is HBM-bound on streaming input_features, so everything is
// fused into one kernel: one lane per batch row for the K=512 reduction and
// the 8-dim statevector circuit, then V_WMMA_F32_16X16X4_F32 for the dense
// 32x100 post-GEMM tile each wave owns (K=4 pads K=3, bias folded into C).
//
// The main kernel is guard-free (full 256-row blocks only); any B%256 tail
// is handled by a tiny scalar kernel so the hot path carries zero branches.
// Transcendentals use the hardware fast path (v_sin/v_cos/v_exp/v_rcp):
// all angles are half-angles bounded by pi/4 + tiny weights, well inside the
// fast-path accuracy envelope for an f32 pipeline.
// ---------------------------------------------------------------------------

typedef __attribute__((ext_vector_type(2)))  float    v2f;
typedef __attribute__((ext_vector_type(8)))  float    v8f;

#define FDIM 512
#define NQ   3
#define NOUT 100
#define NTILES 7   // ceil(100/16)

__device__ __forceinline__ float fast_tanh(float x) {
  x = fminf(fmaxf(x, -15.f), 15.f);      // keep e^{2x} finite
  float e = __expf(2.f * x);             // v_exp_f32
  return __fdividef(e - 1.f, e + 1.f);   // v_rcp_f32 + mul
}

// RY(theta) on the qubit whose state-index bit corresponds to STRIDE
// (qubit0 -> 4, qubit1 -> 2, qubit2 -> 1). Real 2x2 rotation on pairs.
template <int STRIDE>
__device__ __forceinline__ void apply_ry(float (&st)[8], float c, float s) {
#pragma unroll
  for (int base = 0; base < 8; ++base) {
    if ((base & STRIDE) == 0) {
      float x0 = st[base], x1 = st[base + STRIDE];
      st[base]          = c * x0 - s * x1;
      st[base + STRIDE] = s * x0 + c * x1;
    }
  }
}

// Stages 1+2: stream one row (K=512 dot products), tanh squash, 3-qubit
// statevector circuit in registers. Returns <Z0>,<Z1>,<Z2>.
__device__ __forceinline__ void qnet_row(
    const float* __restrict__ input, const float* __restrict__ preW,
    const float* __restrict__ preb, const float* __restrict__ qparams,
    int row, float& z0, float& z1, float& z2)
{
  // Lane streams its own row with b128 loads; preW indices are uniform
  // (loop counter) so those dereferences scalarize to s_load_b128.
  const float4* __restrict__ xr = (const float4*)(input + (size_t)row * FDIM);
  const float4* __restrict__ w0 = (const float4*)(preW);
  const float4* __restrict__ w1 = (const float4*)(preW + FDIM);
  const float4* __restrict__ w2 = (const float4*)(preW + 2 * FDIM);

  float a0 = 0.f, a1 = 0.f, a2 = 0.f;
#pragma unroll 4
  for (int k = 0; k < FDIM / 4; ++k) {
    float4 x  = xr[k];
    float4 p0 = w0[k];
    float4 p1 = w1[k];
    float4 p2 = w2[k];
    a0 += x.x * p0.x + x.y * p0.y + x.z * p0.z + x.w * p0.w;
    a1 += x.x * p1.x + x.y * p1.y + x.z * p1.z + x.w * p1.w;
    a2 += x.x * p2.x + x.y * p2.y + x.z * p2.z + x.w * p2.w;
  }

  const float HPI = 1.5707963267948966f;
  float q0 = fast_tanh(a0 + preb[0]) * HPI;
  float q1 = fast_tanh(a1 + preb[1]) * HPI;
  float q2 = fast_tanh(a2 + preb[2]) * HPI;

  // state index = qubit0*4 + qubit1*2 + qubit2 ; init H|000> = 1/sqrt(8).
  float st[8];
#pragma unroll
  for (int i = 0; i < 8; ++i) st[i] = 0.35355339059327373f;

  apply_ry<4>(st, __cosf(0.5f * q0), __sinf(0.5f * q0));
  apply_ry<2>(st, __cosf(0.5f * q1), __sinf(0.5f * q1));
  apply_ry<1>(st, __cosf(0.5f * q2), __sinf(0.5f * q2));

#pragma unroll
  for (int layer = 1; layer <= 2; ++layer) {
    // CNOT(q0 -> q1): where bit2 set, swap bit1
    float t;
    t = st[4]; st[4] = st[6]; st[6] = t;
    t = st[5]; st[5] = st[7]; st[7] = t;
    // CNOT(q1 -> q2): where bit1 set, swap bit0
    t = st[2]; st[2] = st[3]; st[3] = t;
    t = st[6]; st[6] = st[7]; st[7] = t;
    // uniform (scalar) weight angles for this layer
    float t0 = qparams[layer * NQ + 0];
    float t1 = qparams[layer * NQ + 1];
    float t2 = qparams[layer * NQ + 2];
    apply_ry<4>(st, __cosf(0.5f * t0), __sinf(0.5f * t0));
    apply_ry<2>(st, __cosf(0.5f * t1), __sinf(0.5f * t1));
    apply_ry<1>(st, __cosf(0.5f * t2), __sinf(0.5f * t2));
  }

  float p[8];
#pragma unroll
  for (int i = 0; i < 8; ++i) p[i] = st[i] * st[i];
  z0 = (p[0] + p[1] + p[2] + p[3]) - (p[4] + p[5] + p[6] + p[7]);
  z1 = (p[0] + p[1] + p[4] + p[5]) - (p[2] + p[3] + p[6] + p[7]);
  z2 = (p[0] + p[2] + p[4] + p[6]) - (p[1] + p[3] + p[5] + p[7]);
}

// Main kernel: full 256-row blocks only (no bounds checks anywhere).
__global__ __launch_bounds__(256) void qnet_fused(
    const float* __restrict__ input,   // [B, 512]
    const float* __restrict__ preW,    // [3, 512]
    const float* __restrict__ preb,    // [3]
    const float* __restrict__ qparams, // [45]
    const float* __restrict__ postW,   // [100, 3]
    const float* __restrict__ postb,   // [100]
    float* __restrict__ out)           // [B, 100]
{
  const int tid  = threadIdx.x;
  const int lane = tid & 31;
  const int row  = blockIdx.x * 256 + tid;

  float z0, z1, z2;
  qnet_row(input, preW, preb, qparams, row, z0, z1, z2);

  // ------------------- Stage 3: post-GEMM via WMMA -------------------------
  // Wave owns 32 rows; out tile is 32x100. D = A(16x4) * B(4x16) + C with
  // K=4 padding K=3, bias folded into C. 2 M-tiles x 7 N-tiles = 14 WMMAs.
  const int  lh = lane & 15;
  const bool lo = lane < 16;

  float W0[NTILES], W1[NTILES];
  v8f   Cv[NTILES];
#pragma unroll
  for (int nt = 0; nt < NTILES; ++nt) {
    int n = nt * 16 + lh;
    float v0 = 0.f, v1 = 0.f, v2 = 0.f, bb = 0.f;
    if (n < NOUT) {
      v0 = postW[n * NQ + 0];
      v1 = postW[n * NQ + 1];
      v2 = postW[n * NQ + 2];
      bb = postb[n];
    }
    // B(4x16) layout: lanes 0-15 hold K=0/1 rows, lanes 16-31 hold K=2/3.
    W0[nt] = lo ? v0 : v2;
    W1[nt] = lo ? v1 : 0.f;
#pragma unroll
    for (int r = 0; r < 8; ++r) Cv[nt][r] = bb;    // bias broadcast over M
  }

  const int waveRowBase = row - lane;

#pragma unroll
  for (int t = 0; t < 2; ++t) {
    // Gather the z-vector of row (waveRowBase + t*16 + lh) into A layout:
    // lanes 0-15 hold K=0,1 of row M=lane; lanes 16-31 hold K=2,pad of M=lane-16.
    int   src = t * 16 + lh;
    float z0s = __shfl(z0, src, 32);
    float z1s = __shfl(z1, src, 32);
    float z2s = __shfl(z2, src, 32);

    v2f A;
    A.x = lo ? z0s : z2s;
    A.y = lo ? z1s : 0.f;

    // Row of D this lane's VGPR0 maps to (C/D layout: lanes16-31 => M+8).
    const int mrow0 = waveRowBase + t * 16 + (lo ? 0 : 8);
    // One 64-bit base per (t); stores below use constant byte offsets.
    float* __restrict__ pbase = out + (size_t)mrow0 * NOUT;

#pragma unroll
    for (int nt = 0; nt < NTILES; ++nt) {
      v2f Bv;
      Bv.x = W0[nt];
      Bv.y = W1[nt];

      v8f D = __builtin_amdgcn_wmma_f32_16x16x4_f32(
                  false, A, false, Bv, 0, Cv[nt], false, false);

      const int n = nt * 16 + lh;
      float* __restrict__ pn = pbase + n;
      if ((nt + 1) * 16 <= NOUT) {
        // Compile-time full tile: unconditional clause of 8 stores.
#pragma unroll
        for (int r = 0; r < 8; ++r) pn[r * NOUT] = D[r];
      } else {
        // Last tile (n = 96..111): lane predicate only.
        if (n < NOUT) {
#pragma unroll
          for (int r = 0; r < 8; ++r) pn[r * NOUT] = D[r];
        }
      }
    }
  }
}

// Tail kernel for B % 256 rows (never launched for B = 65536). Kept
// deliberately small: rolled loops, scalar post-GEMM.
__global__ __launch_bounds__(256) void qnet_tail(
    const float* __restrict__ input, const float* __restrict__ preW,
    const float* __restrict__ preb, const float* __restrict__ qparams,
    const float* __restrict__ postW, const float* __restrict__ postb,
    float* __restrict__ out, int rowStart, int B)
{
  const int row = rowStart + blockIdx.x * blockDim.x + threadIdx.x;
  if (row >= B) return;

  float z0, z1, z2;
  qnet_row(input, preW, preb, qparams, row, z0, z1, z2);

  float* __restrict__ po = out + (size_t)row * NOUT;
#pragma unroll 1
  for (int o = 0; o < NOUT; ++o)
    po[o] = z0 * postW[o * NQ + 0] + z1 * postW[o * NQ + 1] +
            z2 * postW[o * NQ + 2] + postb[o];
}

extern "C" void kernel_launch(void* const* d_in, const int* in_sizes, int n_in,
                              void* d_out, int out_size, void* d_ws, size_t ws_size,
                              hipStream_t stream) {
  const float* input   = (const float*)d_in[0];
  const float* preW    = (const float*)d_in[1];
  const float* preb    = (const float*)d_in[2];
  const float* qparams = (const float*)d_in[3];
  const float* postW   = (const float*)d_in[4];
  const float* postb   = (const float*)d_in[5];
  float* out = (float*)d_out;

  const int B        = in_sizes[0] / FDIM;   // 65536
  const int fullBlks = B / 256;              // 256 blocks of 8 waves x 32 rows
  const int tailRows = B - fullBlks * 256;   // 0 for B = 65536

  if (fullBlks > 0) {
    qnet_fused<<<fullBlks, 256, 0, stream>>>(input, preW, preb, qparams,
                                             postW, postb, out);
  }
  if (tailRows > 0) {
    qnet_tail<<<(tailRows + 255) / 256, 256, 0, stream>>>(
        input, preW, preb, qparams, postW, postb, out,
        fullBlks * 256, B);
  }
}